// WDECTLayer_27401891348669
// MI455X (gfx1250) — compile-verified
//
#include <hip/hip_runtime.h>

// ---------------------------------------------------------------------------
// WDECT layer for MI455X (gfx1250, wave32).
//   Phase 0: zero d_out                       (64*32*32 floats)
//   Phase 1: nh[n][t] = (x[n]*w[n]) . v[:,t]  via V_WMMA_F32_16X16X4_F32
//   Phase 2: 307M sigmoids -> LDS [B,S,T] accumulator (256KB, fits in the
//            320KB WGP LDS) via ds_add_f32, then one global fadd flush.
// ---------------------------------------------------------------------------

typedef __attribute__((ext_vector_type(2))) float v2f;
typedef __attribute__((ext_vector_type(8))) float v8f;

#define NT 32   // directions (T)
#define NS 32   // thresholds (S)

__global__ __launch_bounds__(256)
void zero_f32(float* __restrict__ p, int n) {
    int i = blockIdx.x * blockDim.x + threadIdx.x;
    if (i < n) p[i] = 0.0f;
}

// ---------------------------------------------------------------------------
// Phase 1: node heights via WMMA f32 16x16x4 (K=3 padded with a zero K-slot).
// One wave computes a 16-row x 32-col tile of nh (two 16-col WMMAs).
// A layout (32b 16x4): lanes 0-15 M=lane, VGPR0=K0,VGPR1=K1; lanes 16-31 K2,K3.
// B layout (32b 4x16): lanes 0-15 N=lane, VGPR0=K0,VGPR1=K1; lanes 16-31 K2,K3.
// D layout (32b 16x16): lanes 0-15 N=lane, VGPRj = M=j; lanes 16-31 M=8+j.
// ---------------------------------------------------------------------------
__global__ __launch_bounds__(256)
void node_heights_wmma(const float* __restrict__ x,
                       const float* __restrict__ nw,
                       const float* __restrict__ v,   // [3][32] row-major
                       float* __restrict__ nh,        // [N][32]
                       int nN)
{
    const int lane  = threadIdx.x & 31;
    const int wave  = (blockIdx.x * blockDim.x + threadIdx.x) >> 5;
    const int rowBase = wave * 16;
    if (rowBase >= nN) return;                     // wave-uniform exit

    const int m   = lane & 15;
    const int k0  = (lane < 16) ? 0 : 2;
    const int k1  = k0 + 1;                        // 1 or 3 (3 => zero pad)

    // A fragment: row rowBase+m, columns k0/k1 of (x * node_weight)
    int r  = rowBase + m;
    int rc = (r < nN) ? r : (nN - 1);              // clamp; OOB rows unstored
    float w = nw[rc];
    v2f a;
    a.x = x[rc * 3 + k0] * w;
    a.y = (k1 < 3) ? x[rc * 3 + k1] * w : 0.0f;

    v8f d0, d1;
    {
        v8f c = {};
        // B fragment, column tile 0: n = m
        v2f b;
        b.x = v[k0 * NT + m];
        b.y = (k1 < 3) ? v[k1 * NT + m] : 0.0f;
        d0 = __builtin_amdgcn_wmma_f32_16x16x4_f32(false, a, false, b,
                                                   (short)0, c, false, false);
    }
    {
        v8f c = {};
        // B fragment, column tile 1: n = 16 + m
        v2f b;
        b.x = v[k0 * NT + 16 + m];
        b.y = (k1 < 3) ? v[k1 * NT + 16 + m] : 0.0f;
        d1 = __builtin_amdgcn_wmma_f32_16x16x4_f32(false, a, false, b,
                                                   (short)0, c, false, false);
    }

    // Scatter D fragments to nh[row][t]
    const int mOff = (lane < 16) ? 0 : 8;
    #pragma unroll
    for (int j = 0; j < 8; ++j) {
        int row = rowBase + mOff + j;
        if (row < nN) {
            nh[row * NT + m]      = d0[j];
            nh[row * NT + 16 + m] = d1[j];
        }
    }
}

// ---------------------------------------------------------------------------
// Phase 2: fused ECC accumulation. lane = direction t. One simplex per wave.
// sigmoid(500*(lin[s]-h)) = 0.5*tanh(250*(lin[s]-h)) + 0.5   (v_tanh_f32)
// Sign (+nodes, -edges, +faces) folded into the fma coefficients.
// ---------------------------------------------------------------------------
__device__ __forceinline__ float fast_sigmoid_signed(float z250, float halfsign)
{
#if __has_builtin(__builtin_amdgcn_tanhf)
    return fmaf(halfsign, __builtin_amdgcn_tanhf(z250), halfsign);
#else
    // sign * 1/(1+exp2(-2*z250*log2e))
    const float c = 2.8853900817779268f;  // 2*log2(e)
    float e = __builtin_amdgcn_exp2f(-c * z250);
    return (2.0f * halfsign) * __builtin_amdgcn_rcpf(1.0f + e);
#endif
}

__global__ __launch_bounds__(1024)
void ecc_kernel(const float* __restrict__ nh,
                const int*   __restrict__ batch,
                const int*   __restrict__ ei,   // [2][E]
                const float* __restrict__ ew,
                const int*   __restrict__ fc,   // [3][F]
                const float* __restrict__ fw,
                const float* __restrict__ lin,  // [NS], uniform -> s_loads
                float* __restrict__ out,        // [B][NS][NT]
                int nN, int nE, int nF, int nB)
{
    extern __shared__ float acc[];              // nB * NS * NT floats (256KB)
    const int tid  = threadIdx.x;
    const int nAcc = nB * NS * NT;
    for (int i = tid; i < nAcc; i += blockDim.x) acc[i] = 0.0f;
    __syncthreads();

    const int lane   = tid & 31;                // direction t
    const int wavesB = blockDim.x >> 5;
    const int gwave  = blockIdx.x * wavesB + (tid >> 5);
    const int nwaves = gridDim.x * wavesB;

    const int total = nN + nE + nF;
    for (int item = gwave; item < total; item += nwaves) {
        int   b;
        float h, halfsign;
        if (item < nN) {                                       // node: +sigma
            int n = item;
            b = batch[n];
            h = nh[n * NT + lane];
            halfsign = 0.5f;
        } else if (item < nN + nE) {                           // edge: -sigma
            int e  = item - nN;
            int i0 = ei[e], i1 = ei[nE + e];
            b = batch[i0];
            h = fmaxf(nh[i0 * NT + lane], nh[i1 * NT + lane]) * ew[e];
            halfsign = -0.5f;
        } else {                                               // face: +sigma
            int f  = item - nN - nE;
            int i0 = fc[f], i1 = fc[nF + f], i2 = fc[2 * nF + f];
            b = batch[i0];
            h = fmaxf(fmaxf(nh[i0 * NT + lane], nh[i1 * NT + lane]),
                      nh[i2 * NT + lane]) * fw[f];
            halfsign = 0.5f;
        }

        float* base = acc + b * (NS * NT) + lane;  // b wave-uniform; lane = bank
        const float nh250 = -250.0f * h;
        #pragma unroll
        for (int s = 0; s < NS; ++s) {
            float z   = fmaf(250.0f, lin[s], nh250);          // 250*(lin[s]-h)
            float val = fast_sigmoid_signed(z, halfsign);
            atomicAdd(base + s * NT, val);                    // ds_add_f32
        }
    }

    __syncthreads();
    for (int i = tid; i < nAcc; i += blockDim.x)
        unsafeAtomicAdd(&out[i], acc[i]);                     // global fadd
}

// ---------------------------------------------------------------------------
extern "C" void kernel_launch(void* const* d_in, const int* in_sizes, int n_in,
                              void* d_out, int out_size, void* d_ws, size_t ws_size,
                              hipStream_t stream)
{
    const float* x     = (const float*)d_in[0];
    const float* nw    = (const float*)d_in[1];
    const int*   ei    = (const int*)  d_in[2];
    const float* ew    = (const float*)d_in[3];
    const int*   fc    = (const int*)  d_in[4];
    const float* fw    = (const float*)d_in[5];
    const int*   batch = (const int*)  d_in[6];
    const float* v     = (const float*)d_in[7];
    const float* lin   = (const float*)d_in[8];
    float*       out   = (float*)d_out;

    const int nN = in_sizes[1];            // node_weights
    const int nE = in_sizes[3];            // edge_weights
    const int nF = in_sizes[5];            // face_weights
    const int nS = in_sizes[8];            // lin
    const int nB = out_size / (nS * NT);   // 64

    float* nh = (float*)d_ws;              // [nN][32] = 6.4 MB scratch

    // Phase 0: zero the output (harness poisons it).
    zero_f32<<<(out_size + 255) / 256, 256, 0, stream>>>(out, out_size);

    // Phase 1: WMMA projection. 8 waves/block * 16 rows/wave = 128 rows/block.
    const int rowsPerBlock = 8 * 16;
    node_heights_wmma<<<(nN + rowsPerBlock - 1) / rowsPerBlock, 256, 0, stream>>>(
        x, nw, v, nh, nN);

    // Phase 2: fused ECC. One 256KB LDS histogram per workgroup (32 waves/WGP).
    const size_t shmem = (size_t)nB * NS * NT * sizeof(float);
    ecc_kernel<<<256, 1024, shmem, stream>>>(nh, batch, ei, ew, fc, fw, lin,
                                             out, nN, nE, nF, nB);
}